// dgl_agnn_1099511628222
// MI455X (gfx1250) — compile-verified
//
#include <hip/hip_runtime.h>
#include <cstdint>
#include <cmath>

typedef float v2f __attribute__((ext_vector_type(2)));
typedef float v8f __attribute__((ext_vector_type(8)));

// ---------------------------------------------------------------------------
// fill
// ---------------------------------------------------------------------------
__global__ void fill_f32(float* __restrict__ p, float v, int n) {
  int i = blockIdx.x * blockDim.x + threadIdx.x;
  if (i < n) p[i] = v;
}

// ---------------------------------------------------------------------------
// FC1: Y[N,128] = relu(X[N,256] @ W[256,128] + b)   (f32 WMMA 16x16x4)
// block = 256 thr = 8 waves; one 16-row M tile per block, one 16-col N tile
// per wave. X tile staged in LDS with float4 transactions.
// ---------------------------------------------------------------------------
__global__ __launch_bounds__(256)
void fc1_relu_wmma(const float* __restrict__ X, const float* __restrict__ W,
                   const float* __restrict__ bias, float* __restrict__ Y, int N) {
  __shared__ float sx[16 * 256];
  const int m0  = blockIdx.x * 16;
  const int tid = threadIdx.x;

  // stage 16x256 floats = 16x64 float4; 256 threads x 4 iterations
  const float4* Xv = reinterpret_cast<const float4*>(X);
  float4*       sv = reinterpret_cast<float4*>(sx);
#pragma unroll
  for (int i = tid; i < 16 * 64; i += 256) {
    int r = i >> 6;
    int gr = m0 + r; if (gr >= N) gr = N - 1;
    sv[i] = Xv[(size_t)gr * 64 + (i & 63)];
  }
  __syncthreads();

  const int lane = tid & 31;
  const int wv   = tid >> 5;              // 0..7 -> N tile
  const int row  = lane & 15;             // A: M index / B,D: N index
  const int koff = (lane < 16) ? 0 : 2;   // A/B K sub-offset per half-wave
  const int col  = wv * 16 + row;         // 0..127

  v8f acc = {0.f,0.f,0.f,0.f,0.f,0.f,0.f,0.f};
  for (int k0 = 0; k0 < 256; k0 += 4) {
    v2f a, b;
    a.x = sx[row * 256 + k0 + koff];
    a.y = sx[row * 256 + k0 + koff + 1];
    b.x = W[(size_t)(k0 + koff)     * 128 + col];
    b.y = W[(size_t)(k0 + koff + 1) * 128 + col];
    acc = __builtin_amdgcn_wmma_f32_16x16x4_f32(false, a, false, b,
                                                (short)0, acc, false, false);
  }
  const float bv  = bias[col];
  const int mbase = m0 + ((lane < 16) ? 0 : 8);
  if (m0 + 16 <= N) {            // full tile: unguarded stores (hot path)
#pragma unroll
    for (int r = 0; r < 8; ++r) {
      float v = acc[r] + bv;
      Y[(size_t)(mbase + r) * 128 + col] = v > 0.f ? v : 0.f;
    }
  } else {                       // partial tail tile (not taken for N=50000)
#pragma unroll
    for (int r = 0; r < 8; ++r) {
      int mr = mbase + r;
      if (mr < N) {
        float v = acc[r] + bv;
        Y[(size_t)mr * 128 + col] = v > 0.f ? v : 0.f;
      }
    }
  }
}

// ---------------------------------------------------------------------------
// FC2: Y[N,40] = X[N,128] @ W[128,40] + b   (f32 WMMA, 1 wave per tile pair)
// N-dim padded to 48 via masked (non-divergent) B loads -> EXEC stays all-1s.
// ---------------------------------------------------------------------------
__global__ __launch_bounds__(32)
void fc2_wmma(const float* __restrict__ X, const float* __restrict__ W,
              const float* __restrict__ bias, float* __restrict__ Y, int N) {
  const int nt = blockIdx.x % 3;
  const int mt = blockIdx.x / 3;
  const int m0 = mt * 16, n0 = nt * 16;
  const int lane = threadIdx.x;
  const int row  = lane & 15;
  const int koff = (lane < 16) ? 0 : 2;
  const int col  = n0 + row;
  const bool colok = (col < 40);
  const int  bcol  = colok ? col : 39;
  const float mask = colok ? 1.f : 0.f;
  int arow = m0 + row; if (arow >= N) arow = N - 1;

  v8f acc = {0.f,0.f,0.f,0.f,0.f,0.f,0.f,0.f};
  for (int k0 = 0; k0 < 128; k0 += 4) {
    v2f a, b;
    a.x = X[(size_t)arow * 128 + k0 + koff];
    a.y = X[(size_t)arow * 128 + k0 + koff + 1];
    b.x = W[(size_t)(k0 + koff)     * 40 + bcol] * mask;
    b.y = W[(size_t)(k0 + koff + 1) * 40 + bcol] * mask;
    acc = __builtin_amdgcn_wmma_f32_16x16x4_f32(false, a, false, b,
                                                (short)0, acc, false, false);
  }
  if (colok) {
    const float bv  = bias[col];
    const int mbase = m0 + ((lane < 16) ? 0 : 8);
    if (m0 + 16 <= N) {
#pragma unroll
      for (int r = 0; r < 8; ++r)
        Y[(size_t)(mbase + r) * 40 + col] = acc[r] + bv;
    } else {
#pragma unroll
      for (int r = 0; r < 8; ++r) {
        int mr = mbase + r;
        if (mr < N) Y[(size_t)mr * 40 + col] = acc[r] + bv;
      }
    }
  }
}

// ---------------------------------------------------------------------------
// Row-wise L2 normalize: H[i,:] = X[i,:] / max(||X[i,:]||, 1e-12)
// one wave32 per node, float4 per lane (32*4 = 128)
// ---------------------------------------------------------------------------
__global__ __launch_bounds__(256)
void l2norm_rows(const float* __restrict__ X, float* __restrict__ H, int N) {
  int gw   = (blockIdx.x * blockDim.x + threadIdx.x) >> 5;
  int lane = threadIdx.x & 31;
  if (gw >= N) return;
  const float4 v = reinterpret_cast<const float4*>(X + (size_t)gw * 128)[lane];
  float ss = v.x*v.x + v.y*v.y + v.z*v.z + v.w*v.w;
#pragma unroll
  for (int off = 16; off > 0; off >>= 1) ss += __shfl_xor(ss, off, 32);
  const float denom = fmaxf(sqrtf(ss), 1e-12f);
  float4 h;
  h.x = v.x / denom; h.y = v.y / denom; h.z = v.z / denom; h.w = v.w / denom;
  reinterpret_cast<float4*>(H + (size_t)gw * 128)[lane] = h;
}

// ---------------------------------------------------------------------------
// float atomic-max via monotone integer punning (handles +/- and -0)
// ---------------------------------------------------------------------------
__device__ __forceinline__ void atomic_max_f32(float* addr, float v) {
  if (__float_as_uint(v) == 0x80000000u) v = 0.0f;   // canonicalize -0 -> +0
  if (v >= 0.0f) atomicMax(reinterpret_cast<int*>(addr), __float_as_int(v));
  else           atomicMin(reinterpret_cast<unsigned int*>(addr), __float_as_uint(v));
}

// ---------------------------------------------------------------------------
// Edge scores: e = beta * <hn[src], hn[dst]>, fused per-dst segment max.
// one wave32 per edge.
// ---------------------------------------------------------------------------
__global__ __launch_bounds__(256)
void edge_score(const float* __restrict__ H, const int* __restrict__ src,
                const int* __restrict__ dst, const float* __restrict__ betap,
                float* __restrict__ e, float* __restrict__ m, int E) {
  int eid  = (blockIdx.x * blockDim.x + threadIdx.x) >> 5;
  int lane = threadIdx.x & 31;
  if (eid >= E) return;
  const int sn = src[eid], dn = dst[eid];
  const float4 a = reinterpret_cast<const float4*>(H + (size_t)sn * 128)[lane];
  const float4 b = reinterpret_cast<const float4*>(H + (size_t)dn * 128)[lane];
  float d = a.x*b.x + a.y*b.y + a.z*b.z + a.w*b.w;
#pragma unroll
  for (int off = 16; off > 0; off >>= 1) d += __shfl_xor(d, off, 32);
  if (lane == 0) {
    const float ev = betap[0] * d;
    e[eid] = ev;
    atomic_max_f32(m + dn, ev);
  }
}

// ---------------------------------------------------------------------------
// ee = exp(e - m[dst]) (in place), s[dst] += ee.  one thread per edge.
// ---------------------------------------------------------------------------
__global__ __launch_bounds__(256)
void edge_expsum(float* __restrict__ e, const int* __restrict__ dst,
                 const float* __restrict__ m, float* __restrict__ s, int E) {
  int i = blockIdx.x * blockDim.x + threadIdx.x;
  if (i >= E) return;
  const int dn = dst[i];
  const float v = expf(e[i] - m[dn]);
  e[i] = v;
  atomicAdd(s + dn, v);
}

// ---------------------------------------------------------------------------
// Y[dst,:] += coeff * (ee / max(s[dst],eps)) * X[src,:]   (original features)
// one wave32 per edge; 4 f32 atomics per lane.
// ---------------------------------------------------------------------------
__global__ __launch_bounds__(256)
void edge_aggregate(const float* __restrict__ X, const int* __restrict__ src,
                    const int* __restrict__ dst, const float* __restrict__ ee,
                    const float* __restrict__ s, const float* __restrict__ coeffp,
                    int cidx, float* __restrict__ Y, int E) {
  int eid  = (blockIdx.x * blockDim.x + threadIdx.x) >> 5;
  int lane = threadIdx.x & 31;
  if (eid >= E) return;
  const int sn = src[eid], dn = dst[eid];
  const float a = coeffp[cidx] * (ee[eid] / fmaxf(s[dn], 1e-12f));
  const float4 xv = reinterpret_cast<const float4*>(X + (size_t)sn * 128)[lane];
  float* yp = Y + (size_t)dn * 128 + lane * 4;
  atomicAdd(yp + 0, a * xv.x);
  atomicAdd(yp + 1, a * xv.y);
  atomicAdd(yp + 2, a * xv.z);
  atomicAdd(yp + 3, a * xv.w);
}

// ---------------------------------------------------------------------------
// host orchestration
// ---------------------------------------------------------------------------
extern "C" void kernel_launch(void* const* d_in, const int* in_sizes, int n_in,
                              void* d_out, int out_size, void* d_ws, size_t ws_size,
                              hipStream_t stream) {
  const float* X     = (const float*)d_in[0];
  const int*   src1  = (const int*)  d_in[1];
  const int*   dst1  = (const int*)  d_in[2];
  const int*   src2  = (const int*)  d_in[3];
  const int*   dst2  = (const int*)  d_in[4];
  const float* oattn = (const float*)d_in[5];
  const float* W1    = (const float*)d_in[6];
  const float* b1    = (const float*)d_in[7];
  const float* W2    = (const float*)d_in[8];
  const float* b2    = (const float*)d_in[9];
  const float* beta1 = (const float*)d_in[10];
  const float* beta2 = (const float*)d_in[11];
  float* out = (float*)d_out;

  const int N = in_sizes[0] / 256;   // 50000 nodes
  const int E = in_sizes[1];         // 800000 edges per view

  const size_t nd = (size_t)N * 128;
  float* ws = (float*)d_ws;
  float* x1 = ws;                    // FC1 output / layer-2 output (reused)
  float* hn = x1 + nd;               // normalized features
  float* x2 = hn + nd;               // layer-1 output
  float* eb = x2 + nd;               // per-edge score -> exp (in place)
  float* mb = eb + E;                // per-dst running max
  float* sb = mb + N;                // per-dst exp sum
  float* x3 = x1;                    // layer-2 output reuses x1 storage

  const int mtiles = (N + 15) / 16;
  const float NEG_INF = -__builtin_huge_valf();
  const int gFeat = (int)((nd + 255) / 256);
  const int gNode = (N + 255) / 256;
  const int gE    = (E + 255) / 256;
  const int gEw   = (E * 32 + 255) / 256;   // wave-per-edge
  const int gNw   = (N * 32 + 255) / 256;   // wave-per-node

  // FC1 + ReLU (WMMA)
  fc1_relu_wmma<<<mtiles, 256, 0, stream>>>(X, W1, b1, x1, N);

  const int* srcs[2] = {src1, src2};
  const int* dsts[2] = {dst1, dst2};

  for (int layer = 0; layer < 2; ++layer) {
    const float* xin  = (layer == 0) ? x1 : x2;
    float*       yout = (layer == 0) ? x2 : x3;
    const float* beta = (layer == 0) ? beta1 : beta2;

    l2norm_rows<<<gNw, 256, 0, stream>>>(xin, hn, N);
    fill_f32<<<gFeat, 256, 0, stream>>>(yout, 0.f, (int)nd);

    for (int v = 0; v < 2; ++v) {
      fill_f32<<<gNode, 256, 0, stream>>>(mb, NEG_INF, N);
      fill_f32<<<gNode, 256, 0, stream>>>(sb, 0.f, N);
      edge_score    <<<gEw, 256, 0, stream>>>(hn, srcs[v], dsts[v], beta, eb, mb, E);
      edge_expsum   <<<gE,  256, 0, stream>>>(eb, dsts[v], mb, sb, E);
      edge_aggregate<<<gEw, 256, 0, stream>>>(xin, srcs[v], dsts[v], eb, sb,
                                              oattn, v, yout, E);
    }
  }

  // FC2 (WMMA)
  fc2_wmma<<<mtiles * 3, 32, 0, stream>>>(x3, W2, b2, out, N);
}